// LinearCaps_23871428231769
// MI455X (gfx1250) — compile-verified
//
#include <hip/hip_runtime.h>
#include <math.h>

#define IN_CAPS 2401
#define IN_SIZE 8
#define OUT_CAPS 8
#define OUT_SIZE 16
#define BATCH 256
#define KDIM (IN_CAPS * IN_SIZE)    // 19208
#define NDIM (OUT_CAPS * OUT_SIZE)  // 128
#define NKC 16                      // K-split chunks for the s-GEMM
#define CPC 151                     // capsules per chunk: 15*151=2265, last chunk 136

typedef float v2f __attribute__((ext_vector_type(2)));
typedef float v4f __attribute__((ext_vector_type(4)));
typedef float v8f __attribute__((ext_vector_type(8)));

// ---------------------------------------------------------------- init b_ij=0, c_ij=1/8
__global__ void k_init_bc(float* __restrict__ bij, float* __restrict__ cij) {
    int t = blockIdx.x * blockDim.x + threadIdx.x;
    if (t < IN_CAPS * OUT_CAPS) { bij[t] = 0.0f; cij[t] = 0.125f; }
}

// ---------------------------------------------------------------- XT[k][b] = x[b][k]
// LDS-tiled transpose: 8 k-rows x 32 batches per block, coalesced on the write side.
__global__ void k_transpose_x(const float* __restrict__ x, float* __restrict__ xt) {
    __shared__ float tile[8][33];
    int rt = blockIdx.x;            // 0..2400 (8 k's each)
    int bt = blockIdx.y;            // 0..7   (32 b's each)
    int tid = threadIdx.x;          // 0..255
    int tr = tid & 7, tb = tid >> 3;
    tile[tr][tb] = x[(size_t)(bt * 32 + tb) * KDIM + rt * 8 + tr];
    __syncthreads();
    int wb = tid & 31, wr = tid >> 5;
    xt[(size_t)(rt * 8 + wr) * BATCH + bt * 32 + wb] = tile[wr][wb];
}

// ---------------------------------------------------------------- s-GEMM (WMMA f32 16x16x4)
// partial_s[kc][b][n] = sum_{k in chunk kc} x[b][k] * c[i,o] * W[i,o,c,d],  k=i*8+d, n=o*16+c
// One wave per (kc, 16-batch M-tile, o N-tile). 2048 waves.
// Job decode forced into SGPRs so the capsule loop is a scalar loop (no exec-mask idiom;
// EXEC provably stays all-ones for WMMA) and the c_ij coefficient becomes an s_load.
__global__ void k_route_gemm(const float* __restrict__ x, const float* __restrict__ W,
                             const float* __restrict__ cij, float* __restrict__ part) {
    int lane = threadIdx.x & 31;
    int job  = __builtin_amdgcn_readfirstlane(blockIdx.x * 4 + (threadIdx.x >> 5)); // 0..2047
    int kc   = job >> 7;               // 0..15  (SGPR)
    int rem  = job & 127;
    int mt   = rem >> 3;               // batch tile 0..15 (SGPR)
    int ot   = rem & 7;                // output capsule 0..7 (SGPR)
    int ln   = lane & 15;
    int half = lane >> 4;
    int k0   = half * 2;               // per-lane K offset {0,1} vs {2,3}

    int cap0 = kc * CPC;
    int cap1 = min(cap0 + CPC, IN_CAPS);

    const float* xrow = x + (size_t)(mt * 16 + ln) * KDIM;  // A row m = ln
    v8f acc = {};
    for (int i = cap0; i < cap1; ++i) {
        float cs = cij[i * OUT_CAPS + ot];                  // wave-uniform -> scalar load
        const float* xi   = xrow + i * 8;
        const float* wrow = W + (size_t)i * 1024 + ot * 128 + ln * 8;  // W[i][ot][c=ln][*]
#pragma unroll
        for (int g = 0; g < 2; ++g) {                       // two K=4 groups per capsule
            int kb = g * 4 + k0;
            v2f a = *(const v2f*)(xi + kb);                 // x[b][i*8+kb .. +1]
            v2f b = *(const v2f*)(wrow + kb);               // W[i,ot,ln,kb .. +1]
            b *= cs;
            acc = __builtin_amdgcn_wmma_f32_16x16x4_f32(false, a, false, b,
                                                        (short)0, acc, false, false);
        }
    }
    // D layout: VGPR j -> row (j + half*8), col ln
    float* pbase = part + ((size_t)kc * BATCH + mt * 16) * NDIM + ot * 16 + ln;
#pragma unroll
    for (int j = 0; j < 8; ++j)
        pbase[(size_t)(j + half * 8) * NDIM] = acc[j];
}

// ---------------------------------------------------------------- reduce K-chunks + squash
// One block per batch b; thread n sums NKC partials, then 16-lane row reduce for |s|^2.
__global__ void k_reduce_squash(const float* __restrict__ part, float* __restrict__ vT,
                                float* __restrict__ out, int last) {
    int b = blockIdx.x;
    int n = threadIdx.x;  // 0..127, o = n>>4 constant within each 16-lane group
    float s = 0.0f;
#pragma unroll
    for (int kc = 0; kc < NKC; ++kc)
        s += part[((size_t)kc * BATCH + b) * NDIM + n];
    float msq = s * s;
#pragma unroll
    for (int m = 1; m < 16; m <<= 1)
        msq += __shfl_xor(msq, m, 32);
    float mag = sqrtf(msq);
    float v = s * (mag / (1.0f + msq));  // == (msq/(1+msq)) * s/mag
    vT[(size_t)n * BATCH + b] = v;       // transposed for the agreement GEMM
    if (last) out[(size_t)b * NDIM + n] = v;
}

// ---------------------------------------------------------------- agreement (WMMA f32 16x16x4)
// D[m][c] = sum_b XT[i0*8+m][b] * v[b][o*16+c]   (M rows = 2 capsules x 8 d's, K = batch)
// then a[i,o] = (1/B) * sum_{d,c} W[i,o,c,d] * D[(i,d)][c]; b_ij[i,o] += a  (unique writer)
__global__ void k_agree(const float* __restrict__ xt, const float* __restrict__ vT,
                        const float* __restrict__ W, float* __restrict__ bij) {
    int lane = threadIdx.x & 31;
    int job  = __builtin_amdgcn_readfirstlane(blockIdx.x * 4 + (threadIdx.x >> 5)); // 0..9607
    int it2  = job >> 3;               // capsule pair 0..1200 (SGPR)
    int o    = job & 7;                // (SGPR)
    int i0   = it2 * 2;
    int ln   = lane & 15;
    int half = lane >> 4;
    int k0   = half * 2;

    int r = min(i0 * 8 + ln, KDIM - 1);                         // A row m = ln (clamped pad)
    const float* arow = xt + (size_t)r * BATCH;
    const float* brow = vT + (size_t)(o * 16 + ln) * BATCH;     // B col n = ln
    v8f acc = {};
#pragma unroll 4
    for (int kb = 0; kb < BATCH; kb += 4) {
        v2f a = *(const v2f*)(arow + kb + k0);
        v2f b = *(const v2f*)(brow + kb + k0);
        acc = __builtin_amdgcn_wmma_f32_16x16x4_f32(false, a, false, b,
                                                    (short)0, acc, false, false);
    }
    // lane holds D rows m = j + half*8  ->  i = i0+half, d = j, c = ln
    int iW = i0 + half;
    float dot = 0.0f;
    if (iW < IN_CAPS) {
        const float* wp = W + (size_t)iW * 1024 + o * 128 + ln * 8;
        v4f w0 = *(const v4f*)wp;
        v4f w1 = *(const v4f*)(wp + 4);
        dot = acc[0] * w0.x + acc[1] * w0.y + acc[2] * w0.z + acc[3] * w0.w
            + acc[4] * w1.x + acc[5] * w1.y + acc[6] * w1.z + acc[7] * w1.w;
    }
#pragma unroll
    for (int m = 1; m < 16; m <<= 1)
        dot += __shfl_xor(dot, m, 32);   // reduce over c within each 16-lane half
    if (ln == 0 && iW < IN_CAPS)
        bij[iW * OUT_CAPS + o] += dot * (1.0f / BATCH);
}

// ---------------------------------------------------------------- softmax over 8 out-caps
__global__ void k_softmax(const float* __restrict__ bij, float* __restrict__ cij) {
    int i = blockIdx.x * blockDim.x + threadIdx.x;
    if (i >= IN_CAPS) return;
    float v[OUT_CAPS];
    float mx = -1e30f;
#pragma unroll
    for (int o = 0; o < OUT_CAPS; ++o) { v[o] = bij[i * OUT_CAPS + o]; mx = fmaxf(mx, v[o]); }
    float sum = 0.0f;
#pragma unroll
    for (int o = 0; o < OUT_CAPS; ++o) { v[o] = __expf(v[o] - mx); sum += v[o]; }
    float inv = 1.0f / sum;
#pragma unroll
    for (int o = 0; o < OUT_CAPS; ++o) cij[i * OUT_CAPS + o] = v[o] * inv;
}

// ---------------------------------------------------------------- driver
extern "C" void kernel_launch(void* const* d_in, const int* in_sizes, int n_in,
                              void* d_out, int out_size, void* d_ws, size_t ws_size,
                              hipStream_t stream) {
    const float* x = (const float*)d_in[0];  // [256][2401][8]
    const float* W = (const float*)d_in[1];  // [2401][8][16][8]
    float* out = (float*)d_out;              // [256][8][16]

    float* ws   = (float*)d_ws;
    float* XT   = ws;                                   // 19208*256   = 4,917,248 f
    float* part = XT + (size_t)KDIM * BATCH;            // 16*256*128  =   524,288 f
    float* vT   = part + (size_t)NKC * BATCH * NDIM;    // 128*256     =    32,768 f
    float* bij  = vT + (size_t)NDIM * BATCH;            // 19,208 f
    float* cij  = bij + IN_CAPS * OUT_CAPS;             // 19,208 f  (total ~22 MB)

    k_init_bc<<<(IN_CAPS * OUT_CAPS + 255) / 256, 256, 0, stream>>>(bij, cij);
    k_transpose_x<<<dim3(IN_CAPS, 8), 256, 0, stream>>>(x, XT);

    for (int it = 0; it < 3; ++it) {
        k_route_gemm<<<512, 128, 0, stream>>>(x, W, cij, part);
        k_reduce_squash<<<BATCH, NDIM, 0, stream>>>(part, vT, out, it == 2 ? 1 : 0);
        if (it < 2) {
            k_agree<<<2402, 128, 0, stream>>>(XT, vT, W, bij);
            k_softmax<<<(IN_CAPS + 127) / 128, 128, 0, stream>>>(bij, cij);
        }
    }
}